// KuramotoSivashinsky_7705171329354
// MI455X (gfx1250) — compile-verified
//
#include <hip/hip_runtime.h>
#include <stdint.h>

// Kuramoto-Sivashinsky explicit Euler, periodic BCs.
// u0: (64, 2048) f32. Output: (NT+1, 64, 2048) f32 trajectory.
//
// One workgroup per batch row; row state double-buffered in LDS.
// k=4 temporal fusion (ghost zones) per barrier round: 4x fewer barriers /
// LDS round trips at 1.75x redundant VALU (VALU is nowhere near the limit).
// Intermediate slices stored nontemporally from registers; the slice that
// must land in LDS anyway (next round's state) is streamed out with CDNA5
// async LDS->global stores (ASYNCcnt, fire-and-forget).

#define KS_N       2048
#define KS_B       64
#define KS_NT      1000
#define KS_DT      0.01f
#define KS_THREADS 256
#define KS_PTS     8          // KS_N / KS_THREADS
#define KS_K       4          // fused time steps per round
#define KS_ROUNDS  (KS_NT / KS_K)   // 250

typedef float v4f __attribute__((ext_vector_type(4)));

__device__ __forceinline__ float ks_step(float um2, float um1, float u,
                                         float up1, float up2,
                                         float nCdt, float nAdt, float nBdt)
{
    // u_new = u + DT*(-c*u*u_x - alpha*u_xx - beta*u_xxxx), DX = 1
    const float s1     = up1 + um1;
    const float u_x    = (up1 - um1) * 0.5f;
    const float u_xx   = __builtin_fmaf(-2.0f, u, s1);
    const float s2     = up2 + um2;
    const float u_xxxx = __builtin_fmaf(6.0f, u, __builtin_fmaf(-4.0f, s1, s2));
    float r = __builtin_fmaf(nCdt, u * u_x, u);
    r = __builtin_fmaf(nAdt, u_xx, r);
    r = __builtin_fmaf(nBdt, u_xxxx, r);
    return r;
}

__global__ __launch_bounds__(KS_THREADS)
void ks_step_kernel(const float* __restrict__ u0,
                    const float* __restrict__ cp,
                    const float* __restrict__ ap,
                    const float* __restrict__ bp,
                    float* __restrict__ out)
{
    __shared__ __align__(16) float lds[2][KS_N];

    const unsigned row  = blockIdx.x;
    const unsigned tid  = threadIdx.x;
    const unsigned base = tid * KS_PTS;          // 32-byte aligned chunk start

    const float nCdt = -KS_DT * cp[0];
    const float nAdt = -KS_DT * ap[0];
    const float nBdt = -KS_DT * bp[0];

    // ---- init: load u0 chunk, emit trajectory slice 0 (NT), seed LDS buf 0 ----
    {
        const float* src = u0 + (size_t)row * KS_N + base;
        const v4f q0 = *(const v4f*)(src);
        const v4f q1 = *(const v4f*)(src + 4);
        float* d0 = out + (size_t)row * KS_N + base;
        __builtin_nontemporal_store(q0, (v4f*)d0);
        __builtin_nontemporal_store(q1, (v4f*)(d0 + 4));
        *(v4f*)(&lds[0][base])     = q0;
        *(v4f*)(&lds[0][base + 4]) = q1;
    }
    __syncthreads();

    float* cur = lds[0];
    float* nxt = lds[1];

    for (int r = 0; r < KS_ROUNDS; ++r) {
        // ---- window a[0..23] = u^{s}[base-8 .. base+15] (periodic) ----
        float a[24];
#pragma unroll
        for (int q = 0; q < 6; ++q) {
            const v4f m = *(const v4f*)(cur + ((base + 4u * q - 8u) & (KS_N - 1u)));
            a[4 * q + 0] = m.x;
            a[4 * q + 1] = m.y;
            a[4 * q + 2] = m.z;
            a[4 * q + 3] = m.w;
        }

        const size_t sbase = (size_t)(4 * r) * (KS_B * KS_N)
                           + (size_t)row * KS_N + base;

        // ---- advance 4 fused steps; valid window shrinks by 2/side/step ----
#pragma unroll
        for (int t = 1; t <= KS_K; ++t) {
            const int L = 24 - 4 * t;            // 20, 16, 12, 8
            float b[20];
#pragma unroll
            for (int j = 0; j < L; ++j)
                b[j] = ks_step(a[j], a[j + 1], a[j + 2], a[j + 3], a[j + 4],
                               nCdt, nAdt, nBdt);
#pragma unroll
            for (int j = 0; j < L; ++j) a[j] = b[j];

            if (t < KS_K) {
                // own chunk of intermediate slice s+t: a[8-2t .. 15-2t]
                const int j0 = 8 - 2 * t;
                const v4f lo = { a[j0],     a[j0 + 1], a[j0 + 2], a[j0 + 3] };
                const v4f hi = { a[j0 + 4], a[j0 + 5], a[j0 + 6], a[j0 + 7] };
                float* dst = out + sbase + (size_t)t * (KS_B * KS_N);
                __builtin_nontemporal_store(lo, (v4f*)dst);
                __builtin_nontemporal_store(hi, (v4f*)(dst + 4));
            }
        }

        // About to overwrite 'nxt' (our own 32B region): its 2 in-flight async
        // stores were issued 2 rounds ago. Async stores complete FIFO per
        // wave, so asynccnt <= 2 guarantees that pair has drained.
        asm volatile("s_wait_asynccnt 0x2" ::: "memory");

        *(v4f*)(nxt + base)     = (v4f){ a[0], a[1], a[2], a[3] };
        *(v4f*)(nxt + base + 4) = (v4f){ a[4], a[5], a[6], a[7] };

        // s_wait_dscnt 0 + barrier: LDS writes visible workgroup-wide and to
        // the async store engine below.
        __syncthreads();

        // ---- stream slice s+4 straight from LDS (ASYNCcnt, nontemporal) ----
        {
            const float* dst = out + (size_t)(4 * r + 4) * (KS_B * KS_N);
            unsigned voff = (row * KS_N + base) * 4u;
            unsigned l0   = (unsigned)(uintptr_t)(nxt + base); // low 32b = LDS byte addr
            asm volatile("global_store_async_from_lds_b128 %0, %1, %2 th:TH_STORE_NT"
                         :: "v"(voff), "v"(l0), "s"(dst) : "memory");
            asm volatile("global_store_async_from_lds_b128 %0, %1, %2 th:TH_STORE_NT"
                         :: "v"(voff + 16u), "v"(l0 + 16u), "s"(dst) : "memory");
        }

        float* t = cur; cur = nxt; nxt = t;
    }
    // s_endpgm performs an implicit wait-idle, draining remaining async stores.
}

extern "C" void kernel_launch(void* const* d_in, const int* in_sizes, int n_in,
                              void* d_out, int out_size, void* d_ws, size_t ws_size,
                              hipStream_t stream) {
    (void)in_sizes; (void)n_in; (void)out_size; (void)d_ws; (void)ws_size;
    const float* u0 = (const float*)d_in[0];
    const float* c  = (const float*)d_in[1];
    const float* a  = (const float*)d_in[2];
    const float* b  = (const float*)d_in[3];
    float* out = (float*)d_out;

    ks_step_kernel<<<KS_B, KS_THREADS, 0, stream>>>(u0, c, a, b, out);
}